// DigitCaps_52570399703513
// MI455X (gfx1250) — compile-verified
//
#include <hip/hip_runtime.h>

// DigitCaps dynamic routing, fused: u_hat recomputed per pass via
// V_WMMA_F32_16X16X4_F32 (CDNA5), routing state collapsed to cumulative V.
// Sizes fixed by the reference.
#define B_   256
#define R_   1152
#define C_   11
#define I_   8
#define O_   16
#define ITERS_ 3

#define BT      16              // batch tile (WMMA N)
#define RSPLIT  16              // workgroups along R
#define RCHUNK  (R_ / RSPLIT)   // 72 r-values per workgroup
#define NWAVES  C_              // one wave per capsule
#define NTHREADS (NWAVES * 32)  // 352
#define XS_PAD  9               // LDS padding: 9 coprime with 64 banks

typedef __attribute__((ext_vector_type(2))) float v2f;
typedef __attribute__((ext_vector_type(8))) float v8f;

// ---------------------------------------------------------------------------
// Pass kernel: for each (batch-tile, r-slot) compute partial
//   s_part[slot][b][c][o] = sum_{r in slot} softmax_c(u_hat . Vcum) * u_hat
// u_hat built in registers from W,x via two wmma_f32_16x16x4_f32 (K=8).
// ---------------------------------------------------------------------------
__global__ __launch_bounds__(NTHREADS, 1) void caps_pass_kernel(
    const float* __restrict__ x,     // (B,R,I)
    const float* __restrict__ W,     // (R,C,O,I)
    const float* __restrict__ Vcum,  // (B,C,O)  cumulative v from prior iters
    float* __restrict__ s_part)      // (RSPLIT,B,C,O)
{
    __shared__ float xs[RCHUNK][BT][XS_PAD];   // staged x tile
    __shared__ float logits[C_][BT];           // per-r routing logits

    const int wave = threadIdx.x >> 5;         // capsule c = wave id (0..10)
    const int lane = threadIdx.x & 31;
    const int n    = lane & 15;                // batch column within tile
    const int hi   = lane >> 4;                // 0: o=0..7 / K=0..1 ; 1: upper
    const int btile = blockIdx.x & (B_ / BT - 1);
    const int rslot = blockIdx.x >> 4;
    const int b0 = btile * BT;
    const int r0 = rslot * RCHUNK;

    // Cooperative stage of x[b0..b0+15][r0..r0+71][0..7] into LDS.
    for (int idx = threadIdx.x; idx < RCHUNK * BT * I_; idx += NTHREADS) {
        const int i  = idx & 7;
        const int nn = (idx >> 3) & 15;
        const int rl = idx >> 7;
        xs[rl][nn][i] = x[((b0 + nn) * R_ + (r0 + rl)) * I_ + i];
    }

    // Vcum fragment matching the D layout: vc[j] = Vcum[b0+n][wave][j+8*hi]
    float vc[8];
#pragma unroll
    for (int j = 0; j < 8; ++j)
        vc[j] = Vcum[((b0 + n) * C_ + wave) * O_ + 8 * hi + j];

    v8f acc = {};   // partial s: acc[j] ~ s[b0+n][wave][j+8*hi]
    __syncthreads();

    for (int rl = 0; rl < RCHUNK; ++rl) {
        const int r = r0 + rl;

        // A fragment (16x4 f32): lane l holds W[r][c][o=l][k], k = 2*hi+{0,1}
        const float* wp = W + (((r * C_ + wave) * O_ + n) * I_) + 2 * hi;
        v2f a0 = { wp[0], wp[1] };       // K-half 0: k in 0..3
        v2f a1 = { wp[4], wp[5] };       // K-half 1: k in 4..7

        // B fragment (4x16 f32): lane n holds x[b0+n][r][k], k = 2*hi+{0,1}
        const float* xp = &xs[rl][n][2 * hi];
        v2f bb0 = { xp[0], xp[1] };
        v2f bb1 = { xp[4], xp[5] };

        // u[j] = u_hat[b0+n][r][wave][o = j + 8*hi]
        v8f u = {};
        u = __builtin_amdgcn_wmma_f32_16x16x4_f32(false, a0, false, bb0,
                                                  (short)0, u, false, false);
        u = __builtin_amdgcn_wmma_f32_16x16x4_f32(false, a1, false, bb1,
                                                  (short)0, u, false, false);

        // logit[b0+n][r][wave] = sum_o u_hat * Vcum  (fold the two halves)
        float lp = 0.f;
#pragma unroll
        for (int j = 0; j < 8; ++j) lp += u[j] * vc[j];
        lp += __shfl_xor(lp, 16, 32);            // wave32 half-swap
        if (hi == 0) logits[wave][n] = lp;
        __syncthreads();

        // softmax over the 11 capsules for (b0+n, r)
        float mx = logits[0][n];
#pragma unroll
        for (int c = 1; c < C_; ++c) mx = fmaxf(mx, logits[c][n]);
        float se = 0.f;
#pragma unroll
        for (int c = 0; c < C_; ++c) se += __expf(logits[c][n] - mx);
        const float coef = __expf(logits[wave][n] - mx) / se;

#pragma unroll
        for (int j = 0; j < 8; ++j) acc[j] += coef * u[j];
        __syncthreads();   // protect logits before next r overwrites
    }

    // Store this slot's partial s (overwrite; no init needed).
    float* sp = s_part + (((rslot * B_ + (b0 + n)) * C_ + wave) * O_) + 8 * hi;
#pragma unroll
    for (int j = 0; j < 8; ++j) sp[j] = acc[j];
}

// ---------------------------------------------------------------------------
// Reduce partial s over RSPLIT slots, squash, update Vcum (or emit output).
// One thread per (b,c).
// ---------------------------------------------------------------------------
__global__ void caps_squash_kernel(const float* __restrict__ s_part,
                                   float* __restrict__ Vcum,
                                   float* __restrict__ out, int last)
{
    const int idx = blockIdx.x * blockDim.x + threadIdx.x;
    if (idx >= B_ * C_) return;
    const int b = idx / C_;
    const int c = idx % C_;

    float s[O_];
#pragma unroll
    for (int o = 0; o < O_; ++o) s[o] = 0.f;
    for (int slot = 0; slot < RSPLIT; ++slot) {
        const float* sp = s_part + (((slot * B_ + b) * C_ + c) * O_);
#pragma unroll
        for (int o = 0; o < O_; ++o) s[o] += sp[o];
    }

    float sq = 0.f;
#pragma unroll
    for (int o = 0; o < O_; ++o) sq += s[o] * s[o];
    const float scale = (sq / (1.f + sq)) * __frsqrt_rn(sq + 1e-9f);

    if (last) {
        float* op = out + (b * C_ + c) * O_;
#pragma unroll
        for (int o = 0; o < O_; ++o) op[o] = scale * s[o];
    } else {
        float* vp = Vcum + (b * C_ + c) * O_;
#pragma unroll
        for (int o = 0; o < O_; ++o) vp[o] += scale * s[o];
    }
}

__global__ void caps_zero_kernel(float* __restrict__ p, int nelem)
{
    const int idx = blockIdx.x * blockDim.x + threadIdx.x;
    if (idx < nelem) p[idx] = 0.f;
}

// ---------------------------------------------------------------------------
extern "C" void kernel_launch(void* const* d_in, const int* in_sizes, int n_in,
                              void* d_out, int out_size, void* d_ws, size_t ws_size,
                              hipStream_t stream)
{
    const float* x = (const float*)d_in[0];   // (B,R,I)
    const float* W = (const float*)d_in[1];   // (R,C,O,I)
    float* out = (float*)d_out;               // (B,C,O)

    // Workspace layout: s_part (RSPLIT*B*C*O) | Vcum (B*C*O)
    float* s_part = (float*)d_ws;
    float* Vcum   = s_part + (size_t)RSPLIT * B_ * C_ * O_;

    const int nV = B_ * C_ * O_;              // 45056
    caps_zero_kernel<<<(nV + 255) / 256, 256, 0, stream>>>(Vcum, nV);

    const dim3 passGrid((B_ / BT) * RSPLIT);  // 256 workgroups
    const dim3 passBlock(NTHREADS);           // 352 = 11 waves
    const int nBC = B_ * C_;                  // 2816

    for (int it = 0; it < ITERS_; ++it) {
        caps_pass_kernel<<<passGrid, passBlock, 0, stream>>>(x, W, Vcum, s_part);
        caps_squash_kernel<<<(nBC + 255) / 256, 256, 0, stream>>>(
            s_part, Vcum, out, it == ITERS_ - 1 ? 1 : 0);
    }
}